// FCCEncoder_51926154609186
// MI455X (gfx1250) — compile-verified
//
#include <hip/hip_runtime.h>

// ============================================================================
// FCC orientation encoder for MI455X (gfx1250, wave32, WGP).
//
// f_l = Ry_sparse(alpha) * G_l * Ry_sparse(beta) * G_l^T * Ry_sparse(gamma) * s_l
// where G_l = expm(+-pi/2 * K2) conjugates the sparse "vertical axis" rotation
// into the X-axis rotation (so(3) Ad identity). The dense G matvecs are batched
// 16 pixels/column into v_wmma_f32_16x16x4_f32 (full fp32 WMMA). Constants are
// rebuilt on-device every launch by fcc_setup_kernel (deterministic).
// ============================================================================

#define H_DIM 384
#define W_DIM 384
#define NPIX  (H_DIM * W_DIM)
#define LSTR  18   // per-pixel LDS row stride in floats (16 payload + 2 skew, 8B-aligned)

typedef float v2f __attribute__((ext_vector_type(2)));
typedef float v8f __attribute__((ext_vector_type(8)));

#if defined(__HIP_DEVICE_COMPILE__) && defined(__gfx1250__) && \
    __has_builtin(__builtin_amdgcn_wmma_f32_16x16x4_f32)
#define USE_WMMA 1
#else
#define USE_WMMA 0
#endif

#define MAXN 13

// Constant tables (recomputed each launch by fcc_setup_kernel):
//   [   0, 256)  G4  zero-padded to 16x16
//   [ 256, 512)  G4^T
//   [ 512, 768)  G6
//   [ 768,1024)  G6^T
//   [1024,1028)  freq4[m-1] = A4[l+m][l-m], m=1..4  (sparse-rotation frequencies)
//   [1032,1038)  freq6[m-1], m=1..6
__device__ float g_consts[1040];

// --------------------------- setup kernel helpers ---------------------------
__device__ static void mat_mul(const double A[MAXN][MAXN], const double B[MAXN][MAXN],
                               double C[MAXN][MAXN], int n) {
  for (int i = 0; i < n; ++i)
    for (int k = 0; k < n; ++k) {
      double s = 0.0;
      for (int j = 0; j < n; ++j) s += A[i][j] * B[j][k];
      C[i][k] = s;
    }
}

// Out = Re(Q^H X Q), X complex (Xre,Xim), Q complex (Qre,Qim)
__device__ static void conj_to_real(const double Qre[MAXN][MAXN], const double Qim[MAXN][MAXN],
                                    const double Xre[MAXN][MAXN], const double Xim[MAXN][MAXN],
                                    double Out[MAXN][MAXN], int n) {
  double Tre[MAXN][MAXN], Tim[MAXN][MAXN];
  for (int i = 0; i < n; ++i)
    for (int k = 0; k < n; ++k) {
      double tr = 0.0, ti = 0.0;
      for (int j = 0; j < n; ++j) {
        tr += Xre[i][j] * Qre[j][k] - Xim[i][j] * Qim[j][k];
        ti += Xre[i][j] * Qim[j][k] + Xim[i][j] * Qre[j][k];
      }
      Tre[i][k] = tr;
      Tim[i][k] = ti;
    }
  for (int i = 0; i < n; ++i)
    for (int k = 0; k < n; ++k) {
      double s = 0.0;
      for (int j = 0; j < n; ++j)
        s += Qre[j][i] * Tre[j][k] + Qim[j][i] * Tim[j][k];  // Re(conj(Q)*T)
      Out[i][k] = s;
    }
}

// E = expm(M), scaling (1/256) + 14-term Taylor + 8 squarings (f64)
__device__ static void mat_expm(const double M[MAXN][MAXN], double E[MAXN][MAXN], int n) {
  double A[MAXN][MAXN], T[MAXN][MAXN], Tn[MAXN][MAXN];
  const double sc = 1.0 / 256.0;
  for (int i = 0; i < n; ++i)
    for (int k = 0; k < n; ++k) {
      A[i][k] = M[i][k] * sc;
      E[i][k] = (i == k) ? 1.0 : 0.0;
      T[i][k] = (i == k) ? 1.0 : 0.0;
    }
  for (int t = 1; t <= 14; ++t) {
    mat_mul(T, A, Tn, n);
    const double r = 1.0 / (double)t;
    for (int i = 0; i < n; ++i)
      for (int k = 0; k < n; ++k) {
        T[i][k] = Tn[i][k] * r;
        E[i][k] += T[i][k];
      }
  }
  for (int s = 0; s < 8; ++s) {
    mat_mul(E, E, Tn, n);
    for (int i = 0; i < n; ++i)
      for (int k = 0; k < n; ++k) E[i][k] = Tn[i][k];
  }
}

__global__ void __launch_bounds__(32) fcc_setup_kernel() {
  if (threadIdx.x != 0 || blockIdx.x != 0) return;
  for (int li = 0; li < 2; ++li) {
    const int l = li ? 6 : 4;
    const int n = 2 * l + 1;
    const double j = (double)l;

    double hs[MAXN];
    for (int i = 0; i < n - 1; ++i) {
      const double m = -j + (double)i;
      hs[i] = 0.5 * sqrt(j * (j + 1.0) - m * (m + 1.0));
    }

    // Change of basis Q (real<->complex SH). (-i)^l prefactor is real +-1 for
    // even l and cancels inside Q^H X Q, so it is omitted.
    double Qre[MAXN][MAXN], Qim[MAXN][MAXN];
    for (int a = 0; a < MAXN; ++a)
      for (int b = 0; b < MAXN; ++b) { Qre[a][b] = 0.0; Qim[a][b] = 0.0; }
    const double is2 = 0.70710678118654752440;
    for (int m = -l; m < 0; ++m) {
      Qre[l + m][l - m] = is2;
      Qim[l + m][l + m] = -is2;
    }
    Qre[l][l] = 1.0;
    for (int m = 1; m <= l; ++m) {
      const double sg = (m & 1) ? -1.0 : 1.0;
      Qre[l + m][l + m] = sg * is2;
      Qim[l + m][l - m] = sg * is2;
    }

    double Xre[MAXN][MAXN], Xim[MAXN][MAXN];
    double K0[MAXN][MAXN], K1[MAXN][MAXN], K2[MAXN][MAXN];

    // K0 = Re(Q^H X0 Q),  X0 = (raising+lowering)/2 (real antisym tridiag)
    for (int a = 0; a < MAXN; ++a)
      for (int b = 0; b < MAXN; ++b) { Xre[a][b] = 0.0; Xim[a][b] = 0.0; }
    for (int i = 0; i < n - 1; ++i) { Xre[i + 1][i] = -hs[i]; Xre[i][i + 1] = hs[i]; }
    conj_to_real(Qre, Qim, Xre, Xim, K0, n);

    // K1 = Re(Q^H X1 Q),  X1 = diag(i*m)  -> sparse "vertical axis" generator
    for (int a = 0; a < MAXN; ++a)
      for (int b = 0; b < MAXN; ++b) { Xre[a][b] = 0.0; Xim[a][b] = 0.0; }
    for (int i = 0; i < n; ++i) Xim[i][i] = -j + (double)i;
    conj_to_real(Qre, Qim, Xre, Xim, K1, n);

    // K2 = Re(Q^H X2 Q),  X2 = -0.5 i (raising - lowering)
    for (int a = 0; a < MAXN; ++a)
      for (int b = 0; b < MAXN; ++b) { Xre[a][b] = 0.0; Xim[a][b] = 0.0; }
    for (int i = 0; i < n - 1; ++i) { Xim[i + 1][i] = hs[i]; Xim[i][i + 1] = hs[i]; }
    conj_to_real(Qre, Qim, Xre, Xim, K2, n);

    // Orientation sign: [K2, K1] = sgn * K0  (check at the largest K0 entry)
    double C1[MAXN][MAXN], C2[MAXN][MAXN];
    mat_mul(K2, K1, C1, n);
    mat_mul(K1, K2, C2, n);
    double bmag = 0.0, sgn = 1.0;
    for (int a = 0; a < n; ++a)
      for (int b = 0; b < n; ++b) {
        const double v = fabs(K0[a][b]);
        if (v > bmag) {
          bmag = v;
          sgn = ((C1[a][b] - C2[a][b]) * K0[a][b] > 0.0) ? 1.0 : -1.0;
        }
      }

    // G = expm(sgn * pi/2 * K2)  =>  expm(b*K0) = G * expm(b*K1) * G^T
    double SM[MAXN][MAXN], G[MAXN][MAXN];
    const double hpi = 1.57079632679489661923;
    for (int a = 0; a < n; ++a)
      for (int b = 0; b < n; ++b) SM[a][b] = sgn * hpi * K2[a][b];
    mat_expm(SM, G, n);

    float* gp = g_consts + (li ? 512 : 0);
    for (int r = 0; r < 16; ++r)
      for (int c = 0; c < 16; ++c) {
        const bool in = (r < n) && (c < n);
        gp[r * 16 + c]       = in ? (float)G[r][c] : 0.0f;
        gp[256 + r * 16 + c] = in ? (float)G[c][r] : 0.0f;
      }
    float* fr = g_consts + 1024 + li * 8;
    for (int m = 1; m <= l; ++m) fr[m - 1] = (float)K1[l + m][l - m];
  }
}

// ------------------------------ main kernel --------------------------------

// f = Rsp(alpha) * G * Rsp(beta) * G^T * Rsp(gamma) * s, dense parts via WMMA.
// P = this wave's LDS region: 32 rows (pixels) x LSTR floats.
template <int L, int NCH>
__device__ inline void wigner_feat(float alpha, float beta, float gamma,
                                   const float* __restrict__ svec,
                                   const float* __restrict__ Gp,
                                   const float* __restrict__ GTp,
                                   const float* __restrict__ freq,
                                   float* __restrict__ P, int lane,
                                   float* __restrict__ out) {
  constexpr int n = 2 * L + 1;
  float u[n];
#pragma unroll
  for (int k = 0; k < n; ++k) u[k] = svec[k];

  // Sparse axial rotation: 2x2 blocks on (l+m, l-m) with frequency freq[m-1].
  auto rot = [&](float t) {
#pragma unroll
    for (int m = 1; m <= L; ++m) {
      float s, c;
      __sincosf(freq[m - 1] * t, &s, &c);
      const float ui = u[L + m], uj = u[L - m];
      u[L + m] = c * ui + s * uj;
      u[L - m] = c * uj - s * ui;
    }
  };

  const int half = (lane >> 4) & 1;  // lanes 16-31 carry the K+2 / M+8 halves
  const int l15 = lane & 15;

  // Dense y = M * u batched over this wave's 32 pixels (two 16-col groups).
  auto dense = [&](const float* __restrict__ M) {
    // scatter own u (zero-padded to 16) into own LDS row
#pragma unroll
    for (int k = 0; k < 16; ++k)
      P[lane * LSTR + k] = (k < n) ? u[k] : 0.0f;
    __syncthreads();
    float res[2][8];
#if USE_WMMA
#pragma unroll
    for (int grp = 0; grp < 2; ++grp) {
      v8f acc = {0.f, 0.f, 0.f, 0.f, 0.f, 0.f, 0.f, 0.f};
#pragma unroll
      for (int c = 0; c < NCH; ++c) {
        // A (16x4 f32): lanes0-15 K={4c,4c+1}, lanes16-31 K={4c+2,4c+3}, M=lane&15
        const float* ga = M + l15 * 16 + c * 4 + half * 2;
        const v2f av = {ga[0], ga[1]};
        // B (4x16 f32): vgpr0 K=4c(+2 hi half), vgpr1 K=4c+1(+2), N=pixel col
        const float* pb = P + (grp * 16 + l15) * LSTR + c * 4 + half * 2;
        const v2f bv = {pb[0], pb[1]};
        acc = __builtin_amdgcn_wmma_f32_16x16x4_f32(false, av, false, bv,
                                                    (short)0, acc, false, false);
      }
#pragma unroll
      for (int r = 0; r < 8; ++r) res[grp][r] = acc[r];
    }
#else
    // scalar fallback with identical result layout (host pass / missing builtin)
#pragma unroll
    for (int grp = 0; grp < 2; ++grp)
#pragma unroll
      for (int r = 0; r < 8; ++r) {
        const int m = half ? (r + 8) : r;
        float s = 0.0f;
        for (int k = 0; k < 16; ++k)
          s += M[m * 16 + k] * P[(grp * 16 + l15) * LSTR + k];
        res[grp][r] = s;
      }
#endif
    __syncthreads();
    // C (16x16 f32): vgpr r -> row (lanes0-15: M=r, lanes16-31: M=r+8), col=lane&15
#pragma unroll
    for (int grp = 0; grp < 2; ++grp)
#pragma unroll
      for (int r = 0; r < 8; ++r)
        P[(grp * 16 + l15) * LSTR + (half ? r + 8 : r)] = res[grp][r];
    __syncthreads();
#pragma unroll
    for (int k = 0; k < n; ++k) u[k] = P[lane * LSTR + k];
  };

  rot(gamma);
  dense(GTp);
  rot(beta);
  dense(Gp);
  rot(alpha);
#pragma unroll
  for (int k = 0; k < n; ++k) out[k] = u[k];
}

// Cubic-symmetry misorientation: argmax over 24 syms on |w| only, then one
// full quaternion product for the winner.
__device__ inline void misori(const float qc[4], const float qn[4],
                              const float* __restrict__ syms,
                              float& ang, float ax[3]) {
  // q_rel = qn * conj(qc)
  const float rw =  qn[0] * qc[0] + qn[1] * qc[1] + qn[2] * qc[2] + qn[3] * qc[3];
  const float rx = -qn[0] * qc[1] + qn[1] * qc[0] - qn[2] * qc[3] + qn[3] * qc[2];
  const float ry = -qn[0] * qc[2] + qn[1] * qc[3] + qn[2] * qc[0] - qn[3] * qc[1];
  const float rz = -qn[0] * qc[3] - qn[1] * qc[2] + qn[2] * qc[1] + qn[3] * qc[0];

  float bw = -1.0f, b0 = 1.0f, b1 = 0.0f, b2 = 0.0f, b3 = 0.0f;
  for (int i = 0; i < 24; ++i) {
    const float s0 = syms[4 * i + 0];
    const float s1 = syms[4 * i + 1];
    const float s2 = syms[4 * i + 2];
    const float s3 = syms[4 * i + 3];
    const float wv = s0 * rw - s1 * rx - s2 * ry - s3 * rz;
    const float aw = fabsf(wv);
    const bool sel = aw > bw;
    bw = sel ? aw : bw;
    b0 = sel ? s0 : b0;
    b1 = sel ? s1 : b1;
    b2 = sel ? s2 : b2;
    b3 = sel ? s3 : b3;
  }
  float qw = b0 * rw - b1 * rx - b2 * ry - b3 * rz;
  float qx = b0 * rx + b1 * rw + b2 * rz - b3 * ry;
  float qy = b0 * ry - b1 * rz + b2 * rw + b3 * rx;
  float qz = b0 * rz + b1 * ry - b2 * rx + b3 * rw;
  const float fl = (qw < 0.f) ? -1.f : 1.f;
  qw *= fl; qx *= fl; qy *= fl; qz *= fl;
  const float wcl = fminf(1.f, fmaxf(-1.f, qw));
  ang = 2.f * acosf(wcl);
  const float s2c = fmaxf(1.f - wcl * wcl, 0.f);
  const float rinv = (s2c > 1e-12f) ? rsqrtf(s2c) : 0.f;  // 0 -> axis = 0
  ax[0] = qx * rinv;
  ax[1] = qy * rinv;
  ax[2] = qz * rinv;
}

__global__ void __launch_bounds__(256) fcc_encoder_kernel(
    const float* __restrict__ quats,
    const float* __restrict__ s4v,
    const float* __restrict__ s6v,
    const float* __restrict__ syms,
    float* __restrict__ out) {
  __shared__ float Pbuf[8 * 32 * LSTR];  // 8 waves x 32 pixels x LSTR floats
  const int tid = threadIdx.x;
  const int lane = tid & 31;
  float* P = Pbuf + (tid >> 5) * (32 * LSTR);
  const int pix = blockIdx.x * 256 + tid;   // grid is exactly NPIX/256 blocks
  const int row = pix / W_DIM;
  const int col = pix - row * W_DIM;

  const float4 qv = ((const float4*)quats)[pix];
  const float qc[4] = {qv.x, qv.y, qv.z, qv.w};
  const float w = qc[0], x = qc[1], y = qc[2], z = qc[3];

  // rotation matrix entries we need
  const float R00 = 1.f - 2.f * (y * y + z * z);
  const float R01 = 2.f * (x * y - z * w);
  const float R02 = 2.f * (x * z + y * w);
  const float R11 = 1.f - 2.f * (x * x + z * z);
  const float R20 = 2.f * (x * z - y * w);
  const float R21 = 2.f * (y * z + x * w);
  const float R22 = 1.f - 2.f * (x * x + y * y);

  // angles: a = atan2(x0,x2), b = acos(x1) with x = column 1 of R (unit, clipped)
  float c0 = R01, c1 = R11, c2 = R21;
  const float inv = rsqrtf(c0 * c0 + c1 * c1 + c2 * c2);
  c0 = fminf(1.f, fmaxf(-1.f, c0 * inv));
  c1 = fminf(1.f, fmaxf(-1.f, c1 * inv));
  c2 = fminf(1.f, fmaxf(-1.f, c2 * inv));
  const float beta = acosf(c1);
  const float alpha = atan2f(c0, c2);
  float sa, ca;
  __sincosf(alpha, &sa, &ca);
  // R2[0][0] = ca*R00 - sa*R20 ; R2[0][2] = ca*R02 - sa*R22 ; c = atan2(R2[0][2], R2[0][0])
  const float gamma = atan2f(ca * R02 - sa * R22, ca * R00 - sa * R20);

  float f4[9], f6[13];
  wigner_feat<4, 3>(alpha, beta, gamma, s4v, g_consts, g_consts + 256,
                    g_consts + 1024, P, lane, f4);
  wigner_feat<6, 4>(alpha, beta, gamma, s6v, g_consts + 512, g_consts + 768,
                    g_consts + 1032, P, lane, f6);

  // misorientation with wrapped neighbors (edges masked afterwards, as in ref)
  const int colR = (col == W_DIM - 1) ? 0 : col + 1;
  const int rowD = (row == H_DIM - 1) ? 0 : row + 1;
  const float4 qrv = ((const float4*)quats)[row * W_DIM + colR];
  const float4 qdv = ((const float4*)quats)[rowD * W_DIM + col];
  const float qr[4] = {qrv.x, qrv.y, qrv.z, qrv.w};
  const float qd[4] = {qdv.x, qdv.y, qdv.z, qdv.w};

  float angx, angy;
  float axx[3], axy[3];
  misori(qc, qr, syms, angx, axx);
  misori(qc, qd, syms, angy, axy);
  if (col == W_DIM - 1) { angx = 0.f; axx[0] = axx[1] = axx[2] = 0.f; }
  if (row == H_DIM - 1) { angy = 0.f; axy[0] = axy[1] = axy[2] = 0.f; }

  // outputs, tuple-concatenated: f0 (N), f1 (N,3), f4 (N,9), f6 (N,13)
  out[pix] = 0.5f * (angx + angy);
  float* f1p = out + NPIX + pix * 3;
  f1p[0] = 0.5f * (axx[0] + axy[0]);
  f1p[1] = 0.5f * (axx[1] + axy[1]);
  f1p[2] = 0.5f * (axx[2] + axy[2]);
  float* f4p = out + 4 * NPIX + (size_t)pix * 9;
#pragma unroll
  for (int k = 0; k < 9; ++k) f4p[k] = f4[k];
  float* f6p = out + 13 * NPIX + (size_t)pix * 13;
#pragma unroll
  for (int k = 0; k < 13; ++k) f6p[k] = f6[k];
}

extern "C" void kernel_launch(void* const* d_in, const int* in_sizes, int n_in,
                              void* d_out, int out_size, void* d_ws, size_t ws_size,
                              hipStream_t stream) {
  (void)in_sizes; (void)n_in; (void)out_size; (void)d_ws; (void)ws_size;
  const float* quats = (const float*)d_in[0];
  const float* s4v   = (const float*)d_in[1];
  const float* s6v   = (const float*)d_in[2];
  const float* syms  = (const float*)d_in[3];
  // H, W are compile-time constants (384) per the reference.
  fcc_setup_kernel<<<1, 32, 0, stream>>>();
  fcc_encoder_kernel<<<NPIX / 256, 256, 0, stream>>>(quats, s4v, s6v, syms,
                                                     (float*)d_out);
}